// SparseMultiHeadDiffAttention_17849884082435
// MI455X (gfx1250) — compile-verified
//
#include <hip/hip_runtime.h>
#include <math.h>

typedef __attribute__((ext_vector_type(2))) float v2f;
typedef __attribute__((ext_vector_type(8))) float v8f;

#define DIM 256
#define HEADS 8
#define EDIM 6
#define LAMBDA_INIT 0.6321205588285577f  // 1 - exp(-1)

// ---------- order-preserving float<->uint encoding for atomicMax ----------
// encode: all finite floats map to u > 0, monotonically; memset-0 is identity.
__device__ __forceinline__ unsigned enc_ordered(float f) {
  unsigned u = __float_as_uint(f);
  return (u & 0x80000000u) ? ~u : (u | 0x80000000u);
}
__device__ __forceinline__ float dec_ordered(unsigned u) {
  unsigned b = (u & 0x80000000u) ? (u & 0x7FFFFFFFu) : ~u;
  return __uint_as_float(b);
}

// ---------------------------------------------------------------------------
// beta = sigmoid(exp(sum(lq*lk)) * LAMBDA_INIT); tiny, single thread.
// ---------------------------------------------------------------------------
__global__ void beta_kernel(const float* __restrict__ lq,
                            const float* __restrict__ lk, int n,
                            float* __restrict__ beta_ws,
                            float* __restrict__ beta_out) {
  if (threadIdx.x == 0 && blockIdx.x == 0) {
    float s = 0.0f;
    for (int i = 0; i < n; ++i) s = fmaf(lq[i], lk[i], s);
    float l1 = expf(s);
    float b = 1.0f / (1.0f + expf(-l1 * LAMBDA_INIT));
    *beta_ws = b;
    *beta_out = b;
  }
}

// ---------------------------------------------------------------------------
// C[M x 256] = A[M x 256] @ W[256 x 256]^T  via V_WMMA_F32_16X16X4_F32.
// One block = 4 waves; block handles one 16-row M tile, each wave a 64-col
// strip (4 N-tiles), reusing the A fragment 4x per K step.
// f32 16x4 A layout: row = M0 + lane%16, cols = k0 + (lane>>4)*2 + {0,1}.
// f32 4x16 B layout is the mirror image -> same addressing over W (N-major).
// C/D layout: VGPR v -> M = v + (lane>>4)*8, N = lane%16.
// Requires M % 16 == 0 (true: 50000 = 3125*16).
// ---------------------------------------------------------------------------
__global__ __launch_bounds__(128) void gemm_nt_wmma(
    const float* __restrict__ A, const float* __restrict__ W,
    float* __restrict__ C, int M) {
  const int lane = threadIdx.x & 31;
  const int wave = threadIdx.x >> 5;        // 0..3
  const int m0 = blockIdx.x << 4;
  const int n0 = wave << 6;                 // 64 columns per wave
  const int r = lane & 15;
  const int koff = (lane >> 4) << 1;        // 0 or 2
  (void)M;

  const float* a_ptr  = A + (size_t)(m0 + r) * DIM + koff;
  const float* w_ptr0 = W + (size_t)(n0 + r) * DIM + koff;
  const float* w_ptr1 = w_ptr0 + (size_t)16 * DIM;
  const float* w_ptr2 = w_ptr0 + (size_t)32 * DIM;
  const float* w_ptr3 = w_ptr0 + (size_t)48 * DIM;

  v8f acc0 = {}; v8f acc1 = {}; v8f acc2 = {}; v8f acc3 = {};
  #pragma unroll 4
  for (int k = 0; k < DIM; k += 4) {
    v2f a  = *(const v2f*)(a_ptr  + k);
    v2f b0 = *(const v2f*)(w_ptr0 + k);
    v2f b1 = *(const v2f*)(w_ptr1 + k);
    v2f b2 = *(const v2f*)(w_ptr2 + k);
    v2f b3 = *(const v2f*)(w_ptr3 + k);
    acc0 = __builtin_amdgcn_wmma_f32_16x16x4_f32(false, a, false, b0, (short)0, acc0, false, false);
    acc1 = __builtin_amdgcn_wmma_f32_16x16x4_f32(false, a, false, b1, (short)0, acc1, false, false);
    acc2 = __builtin_amdgcn_wmma_f32_16x16x4_f32(false, a, false, b2, (short)0, acc2, false, false);
    acc3 = __builtin_amdgcn_wmma_f32_16x16x4_f32(false, a, false, b3, (short)0, acc3, false, false);
  }

  float* c0 = C + (size_t)(m0 + ((lane >> 4) << 3)) * DIM + n0 + r;
  #pragma unroll
  for (int v = 0; v < 8; ++v) {
    c0[(size_t)v * DIM +  0] = acc0[v];
    c0[(size_t)v * DIM + 16] = acc1[v];
    c0[(size_t)v * DIM + 32] = acc2[v];
    c0[(size_t)v * DIM + 48] = acc3[v];
  }
}

// ---------------------------------------------------------------------------
// Edge MLP: h = GELU(edge_attr @ e_w1^T + e_b1) ; scores = h @ e_w2^T + e_b2.
// Weights (~23KB) live in LDS; one thread per edge; lockstep j -> broadcast
// LDS reads (no bank conflicts). Also does pass 1 of the segment softmax:
// atomicMax of ordered-encoded scores into gmax[tgt*16 + slot].
// ---------------------------------------------------------------------------
__global__ __launch_bounds__(256) void edge_mlp_kernel(
    const float* __restrict__ edge_attr,
    const float* __restrict__ e_w1, const float* __restrict__ e_b1,
    const float* __restrict__ e_w2, const float* __restrict__ e_b2,
    const int* __restrict__ tgt,
    float* __restrict__ scores, unsigned* __restrict__ gmax, int E) {
  __shared__ float w1[DIM * EDIM];
  __shared__ float b1[DIM];
  __shared__ float w2[16 * DIM];
  __shared__ float b2s[16];
  for (int i = threadIdx.x; i < DIM * EDIM; i += 256) w1[i] = e_w1[i];
  for (int i = threadIdx.x; i < DIM; i += 256) b1[i] = e_b1[i];
  for (int i = threadIdx.x; i < 16 * DIM; i += 256) w2[i] = e_w2[i];
  if (threadIdx.x < 16) b2s[threadIdx.x] = e_b2[threadIdx.x];
  __syncthreads();

  const int e = blockIdx.x * 256 + threadIdx.x;
  if (e >= E) return;

  float a[EDIM];
  #pragma unroll
  for (int k = 0; k < EDIM; ++k) a[k] = edge_attr[(size_t)e * EDIM + k];

  float s[16];
  #pragma unroll
  for (int o = 0; o < 16; ++o) s[o] = b2s[o];

  for (int j = 0; j < DIM; ++j) {
    float acc = b1[j];
    #pragma unroll
    for (int k = 0; k < EDIM; ++k) acc = fmaf(a[k], w1[j * EDIM + k], acc);
    // exact GELU: 0.5*x*(1+erf(x/sqrt(2)))
    float h = 0.5f * acc * (1.0f + erff(acc * 0.7071067811865476f));
    #pragma unroll
    for (int o = 0; o < 16; ++o) s[o] = fmaf(h, w2[o * DIM + j], s[o]);
  }

  const int t = tgt[e];
  #pragma unroll
  for (int o = 0; o < 16; ++o) {
    scores[(size_t)e * 16 + o] = s[o];
    atomicMax(&gmax[(size_t)t * 16 + o], enc_ordered(s[o]));
  }
}

// ---------------------------------------------------------------------------
// Pass 2: e = exp(score - segment_max); overwrite scores; atomicAdd sums.
// One thread per (edge, slot) of the 16 = H*2 channels.
// ---------------------------------------------------------------------------
__global__ __launch_bounds__(256) void softmax_exp_kernel(
    float* __restrict__ scores, const unsigned* __restrict__ gmax,
    float* __restrict__ gsum, const int* __restrict__ tgt, int E) {
  size_t gid = (size_t)blockIdx.x * 256 + threadIdx.x;
  if (gid >= (size_t)E * 16) return;
  const int e = (int)(gid >> 4);
  const int slot = (int)(gid & 15);
  const int t = tgt[e];
  const float m = dec_ordered(gmax[(size_t)t * 16 + slot]);
  const float ex = expf(scores[gid] - m);
  scores[gid] = ex;
  atomicAdd(&gsum[(size_t)t * 16 + slot], ex);
}

// ---------------------------------------------------------------------------
// Pass 3: attn[e,h] = e1/s1 - beta*(e2/s2), written straight to d_out region.
// ---------------------------------------------------------------------------
__global__ __launch_bounds__(256) void attn_kernel(
    const float* __restrict__ scores, const float* __restrict__ gsum,
    const int* __restrict__ tgt, const float* __restrict__ beta_ws,
    float* __restrict__ attn_out, int E) {
  size_t gid = (size_t)blockIdx.x * 256 + threadIdx.x;
  if (gid >= (size_t)E * HEADS) return;
  const int e = (int)(gid >> 3);
  const int h = (int)(gid & 7);
  const int t = tgt[e];
  const float beta = *beta_ws;
  const float e1 = scores[(size_t)e * 16 + 2 * h];
  const float e2 = scores[(size_t)e * 16 + 2 * h + 1];
  const float s1 = gsum[(size_t)t * 16 + 2 * h];
  const float s2 = gsum[(size_t)t * 16 + 2 * h + 1];
  attn_out[gid] = e1 / s1 - beta * (e2 / s2);
}

// ---------------------------------------------------------------------------
// Weighted scatter: accum[tgt, :] += attn[e,h] * V[src, :]. One block per
// edge, 256 threads = full 256-dim row (thread t -> head t>>5, channel t&31).
// accum (51MB) is L2-resident (192MB) so fp32 atomics resolve in L2.
// ---------------------------------------------------------------------------
__global__ __launch_bounds__(256) void scatter_kernel(
    const float* __restrict__ V, const float* __restrict__ attn,
    const int* __restrict__ src, const int* __restrict__ tgt,
    float* __restrict__ accum, int E) {
  const int e = blockIdx.x;
  const int t = threadIdx.x;
  const int h = t >> 5;
  const float w = attn[(size_t)e * HEADS + h];
  const int sn = src[e];
  const int dn = tgt[e];
  atomicAdd(&accum[(size_t)dn * DIM + t], w * V[(size_t)sn * DIM + t]);
}

// ---------------------------------------------------------------------------
extern "C" void kernel_launch(void* const* d_in, const int* in_sizes, int n_in,
                              void* d_out, int out_size, void* d_ws, size_t ws_size,
                              hipStream_t stream) {
  (void)n_in; (void)out_size; (void)ws_size;
  const float* x         = (const float*)d_in[0];
  const float* edge_attr = (const float*)d_in[1];
  const float* v_w       = (const float*)d_in[2];
  const float* out_w     = (const float*)d_in[3];
  const float* e_w1      = (const float*)d_in[4];
  const float* e_b1      = (const float*)d_in[5];
  const float* e_w2      = (const float*)d_in[6];
  const float* e_b2      = (const float*)d_in[7];
  const float* lq        = (const float*)d_in[8];
  const float* lk        = (const float*)d_in[9];
  const int*   ei        = (const int*)d_in[10];

  const int N = in_sizes[0] / DIM;       // 50000 (multiple of 16)
  const int E = in_sizes[10] / 2;        // 800000
  const int nlam = in_sizes[8];          // 100
  const int* src = ei;
  const int* tgt = ei + E;

  // output layout: out [N*D] | attn [E*H] | beta [1]
  float* out      = (float*)d_out;
  float* attn_out = out + (size_t)N * DIM;
  float* beta_out = attn_out + (size_t)E * HEADS;

  // workspace layout: V | scores | accum | gmax | gsum | beta
  float* wsf      = (float*)d_ws;
  float* V        = wsf;                                   // N*256
  float* scores   = V + (size_t)N * DIM;                   // E*16
  float* accum    = scores + (size_t)E * 16;               // N*256
  unsigned* gmax  = (unsigned*)(accum + (size_t)N * DIM);  // N*16
  float* gsum     = (float*)(gmax + (size_t)N * 16);       // N*16
  float* beta_ws  = gsum + (size_t)N * 16;                 // 1

  // zero accum+gmax+gsum in one contiguous memset (0 == ordered-max identity)
  size_t zero_bytes = ((size_t)N * DIM + (size_t)N * 16 * 2) * sizeof(float);
  hipMemsetAsync(accum, 0, zero_bytes, stream);

  beta_kernel<<<1, 64, 0, stream>>>(lq, lk, nlam, beta_ws, beta_out);

  gemm_nt_wmma<<<N / 16, 128, 0, stream>>>(x, v_w, V, N);

  edge_mlp_kernel<<<(E + 255) / 256, 256, 0, stream>>>(
      edge_attr, e_w1, e_b1, e_w2, e_b2, tgt, scores, gmax, E);

  softmax_exp_kernel<<<(unsigned)(((size_t)E * 16 + 255) / 256), 256, 0, stream>>>(
      scores, gmax, gsum, tgt, E);

  attn_kernel<<<(unsigned)(((size_t)E * HEADS + 255) / 256), 256, 0, stream>>>(
      scores, gsum, tgt, beta_ws, attn_out, E);

  scatter_kernel<<<E, 256, 0, stream>>>(V, attn_out, src, tgt, accum, E);

  gemm_nt_wmma<<<N / 16, 128, 0, stream>>>(accum, out_w, out, N);
}